// TFN_14688788152659
// MI455X (gfx1250) — compile-verified
//
#include <hip/hip_runtime.h>
#include <cstdint>

// ---------------------------------------------------------------------------
// TFN forward for MI455X / gfx1250.
// Dominant cost: [64 x 912673] x [912673 x 96] GEMM where the K matrix (W1,
// ~350 MB fp32) is streamed once from HBM (roofline ~15 us @ 23.3 TB/s).
// Strategy: split-K across 512 workgroups; W1 K-slabs DMA'd into LDS by the
// Tensor Data Mover (double buffered, s_wait_tensorcnt); A ("fusion") tiles
// synthesized on the fly from av=a1(x)v1 and t1; fp32 WMMA 16x16x4 so W1
// bytes feed the matrix unit with no conversion pass.
// ---------------------------------------------------------------------------

#define B_SZ   64
#define H_SZ   96
#define HP1    97
#define K_TOT  (HP1 * HP1 * HP1)                 // 912673
#define SPLITS 512
#define KPW    ((K_TOT + SPLITS - 1) / SPLITS)   // 1783 rows of W1 per WG
#define KS     96                                 // K-slab rows staged in LDS
#define FS_STR (KS + 2)                           // padded fusion-slab stride

typedef float v2f  __attribute__((ext_vector_type(2)));
typedef float v8f  __attribute__((ext_vector_type(8)));
typedef unsigned int u32x4 __attribute__((ext_vector_type(4)));
typedef int   i32x4 __attribute__((ext_vector_type(4)));
typedef int   i32x8 __attribute__((ext_vector_type(8)));

__device__ __forceinline__ int imin(int a, int b) { return a < b ? a : b; }

// --------------------------- TDM 2-D slab load -----------------------------
// Loads a [KS x 96] fp32 tile of W1 (row stride 96 elements) into LDS.
// tensor_dim1 = rows_valid < tile_dim1(KS) => TDM zero-fills the tail rows,
// which makes the ragged last K-slab safe with no extra masking.
__device__ __forceinline__ void tdm_load_w1_slab(float* lds_dst,
                                                 const float* gsrc,
                                                 int rows_valid) {
  uint64_t ga       = (uint64_t)(uintptr_t)gsrc;
  uint32_t lds_addr = (uint32_t)(uintptr_t)lds_dst;   // low 32 bits = LDS byte offset

  u32x4 g0;
  g0[0] = 1u;                                              // count = 1 (valid user D#)
  g0[1] = lds_addr;                                        // lds_addr
  g0[2] = (uint32_t)ga;                                    // global_addr[31:0]
  g0[3] = ((uint32_t)(ga >> 32) & 0x01FFFFFFu) | (2u << 30); // addr[56:32] | type=2

  i32x8 g1;
  g1[0] = (2 << 16);                                       // data_size = 4B; no pad/iterate/mask
  g1[1] = (int)((uint32_t)(H_SZ & 0xFFFF) << 16);          // tensor_dim0[15:0] @ bits[63:48]
  g1[2] = (int)(((uint32_t)rows_valid & 0xFFFFu) << 16);   // dim0 hi=0 | tensor_dim1 lo
  g1[3] = (int)(((uint32_t)H_SZ << 16) |
                (((uint32_t)rows_valid >> 16) & 0xFFFFu)); // tensor_dim1 hi | tile_dim0=96
  g1[4] = KS;                                              // tile_dim1 = KS, tile_dim2 = 0
  g1[5] = H_SZ;                                            // tensor_dim0_stride = 96
  g1[6] = 0;
  g1[7] = 0;

  i32x4 z4 = {0, 0, 0, 0};                                 // 2-D tensor: groups 2/3 unused
  i32x8 z8 = {0, 0, 0, 0, 0, 0, 0, 0};
  __builtin_amdgcn_tensor_load_to_lds(g0, g1, z4, z4, z8, 0);
}

// --------------------------- K1: modality encoders -------------------------
// out[mod][b][j] : j==0 -> 1.0 (the concatenated ones column), else
// relu(x[b,:] . W[:,j-1] + bias[j-1]).  ILP-4 accumulators break the serial
// FMA dependency chain (Kd up to 1024 deep otherwise).
__global__ __launch_bounds__(256) void tfn_encode(
    const float* __restrict__ audios, const float* __restrict__ texts,
    const float* __restrict__ videos,
    const float* __restrict__ Wa, const float* __restrict__ ba,
    const float* __restrict__ Wt, const float* __restrict__ bt,
    const float* __restrict__ Wv, const float* __restrict__ bv,
    float* __restrict__ a1, float* __restrict__ t1, float* __restrict__ v1) {
  int gid = blockIdx.x * 256 + threadIdx.x;
  if (gid >= 3 * B_SZ * HP1) return;
  int mod = gid / (B_SZ * HP1);
  int rem = gid - mod * (B_SZ * HP1);
  int b = rem / HP1;
  int j = rem - b * HP1;

  const float *x, *W, *bias;
  float* out;
  int Kd;
  if (mod == 0)      { x = audios; W = Wa; bias = ba; Kd = 512;  out = a1; }
  else if (mod == 1) { x = texts;  W = Wt; bias = bt; Kd = 1024; out = t1; }
  else               { x = videos; W = Wv; bias = bv; Kd = 512;  out = v1; }

  float r = 1.0f;
  if (j > 0) {
    int h = j - 1;
    const float* xr = x + b * Kd;
    float s0 = 0.f, s1 = 0.f, s2 = 0.f, s3 = 0.f;
    for (int k = 0; k < Kd; k += 4) {            // Kd % 4 == 0
      s0 = fmaf(xr[k + 0], W[(k + 0) * H_SZ + h], s0);
      s1 = fmaf(xr[k + 1], W[(k + 1) * H_SZ + h], s1);
      s2 = fmaf(xr[k + 2], W[(k + 2) * H_SZ + h], s2);
      s3 = fmaf(xr[k + 3], W[(k + 3) * H_SZ + h], s3);
    }
    float s = bias[h] + ((s0 + s1) + (s2 + s3));
    r = fmaxf(s, 0.0f);
  }
  out[b * HP1 + j] = r;
}

// ------------------- K2: av[b, a*97+v] = a1[b,a] * v1[b,v] -----------------
__global__ __launch_bounds__(256) void tfn_av(const float* __restrict__ a1,
                                              const float* __restrict__ v1,
                                              float* __restrict__ av) {
  int e = blockIdx.x * 256 + threadIdx.x;
  if (e >= B_SZ * HP1 * HP1) return;
  int b = e / (HP1 * HP1);
  int r = e - b * (HP1 * HP1);
  int a = r / HP1;
  int v = r - a * HP1;
  av[e] = a1[b * HP1 + a] * v1[b * HP1 + v];
}

// ---------------- K3: split-K fusion GEMM (WMMA f32 16x16x4) ---------------
// Each WG owns K rows [kbeg, kend) of W1, all of M=64, N=96.
// 8 waves: wave w -> M-tile (w&3), N-tiles {0,1,2} or {3,4,5} (w>>2).
__global__ __launch_bounds__(256) void tfn_gemm(const float* __restrict__ W1,
                                                const float* __restrict__ av,
                                                const float* __restrict__ t1,
                                                float* __restrict__ part) {
  __shared__ float wslab[2][KS * H_SZ];       // W1 K-slabs (double buffered, TDM dest)
  __shared__ float fslab[B_SZ * FS_STR];      // fusion A-slab [64 x KS], padded stride

  const int tid  = threadIdx.x;
  const int wave = tid >> 5;
  const int lane = tid & 31;
  const int mt   = wave & 3;      // M-tile 0..3  (rows 16*mt .. 16*mt+15)
  const int nh   = wave >> 2;     // N half: tiles nh*3 .. nh*3+2

  const int split = blockIdx.x;
  const int kbeg  = split * KPW;
  const int kend  = imin(kbeg + KPW, K_TOT);
  const int nslab = (kend - kbeg + KS - 1) / KS;

  v8f acc0 = {}, acc1 = {}, acc2 = {};

  // Prologue: wave 0 launches the first TDM slab load.
  if (wave == 0) {
    tdm_load_w1_slab(&wslab[0][0], W1 + (size_t)kbeg * H_SZ,
                     imin(KS, kend - kbeg));
  }

  int buf = 0;
  for (int s = 0; s < nslab; ++s) {
    const int kbase = kbeg + s * KS;
    const int R     = imin(KS, kend - kbase);

    // Build fusion slab: fslab[b][kk] = av[b, k/97] * t1[b, k%97], zero tail.
    for (int idx = tid; idx < B_SZ * KS; idx += 256) {
      int b  = idx / KS;
      int kk = idx - b * KS;
      float val = 0.0f;
      if (kk < R) {
        int k = kbase + kk;
        int u = k / HP1;            // a*97 + v
        int t = k - u * HP1;        // t index
        val = av[b * (HP1 * HP1) + u] * t1[b * HP1 + t];
      }
      fslab[b * FS_STR + kk] = val;
    }

    if (wave == 0) __builtin_amdgcn_s_wait_tensorcnt(0);  // slab s landed
    __syncthreads();                                      // wslab[buf] + fslab ready

    // Prefetch next slab into the other buffer while we compute.
    if (wave == 0 && (s + 1) < nslab) {
      int kb2 = kbase + KS;
      tdm_load_w1_slab(&wslab[buf ^ 1][0], W1 + (size_t)kb2 * H_SZ,
                       imin(KS, kend - kb2));
    }

    const float* Wb = &wslab[buf][0];
    const int m   = lane & 15;
    const int h2  = (lane >> 4) << 1;            // 0 (lanes 0-15) or 2 (lanes 16-31)
    const int n0  = nh * 48 + (lane & 15);
    const float* fr = &fslab[(mt * 16 + m) * FS_STR];

    for (int kk = 0; kk < KS; kk += 4) {
      // A fragment: lane holds fusion[m][kk+2h .. kk+2h+1]  (ds_load_b64)
      v2f a = *(const v2f*)&fr[kk + h2];
      const float* bp = Wb + (kk + h2) * H_SZ + n0;
      v2f b0 = { bp[0],  bp[H_SZ] };             // K rows kk+2h, kk+2h+1
      v2f b1 = { bp[16], bp[16 + H_SZ] };
      v2f b2 = { bp[32], bp[32 + H_SZ] };
      acc0 = __builtin_amdgcn_wmma_f32_16x16x4_f32(false, a, false, b0,
                                                   (short)0, acc0, false, false);
      acc1 = __builtin_amdgcn_wmma_f32_16x16x4_f32(false, a, false, b1,
                                                   (short)0, acc1, false, false);
      acc2 = __builtin_amdgcn_wmma_f32_16x16x4_f32(false, a, false, b2,
                                                   (short)0, acc2, false, false);
    }
    __syncthreads();   // all reads of fslab/wslab[buf] done before next iter
    buf ^= 1;
  }

  // Write this split's partial [64 x 96] block. C layout: VGPR r, half h:
  // M = 16*mt + r + 8*h, N = 48*nh + 16*ntile + (lane&15).
  float* outp = part + (size_t)split * (B_SZ * H_SZ);
  const int hhalf = lane >> 4;
  const int n0 = nh * 48 + (lane & 15);
#pragma unroll
  for (int r = 0; r < 8; ++r) {
    int M = mt * 16 + r + 8 * hhalf;
    outp[M * H_SZ + n0]      = acc0[r];
    outp[M * H_SZ + n0 + 16] = acc1[r];
    outp[M * H_SZ + n0 + 32] = acc2[r];
  }
}

// ------------- K4: deterministic split-K reduction + bias + relu -----------
__global__ __launch_bounds__(256) void tfn_reduce(const float* __restrict__ part,
                                                  const float* __restrict__ b1,
                                                  float* __restrict__ y1) {
  int e = blockIdx.x * 256 + threadIdx.x;
  if (e >= B_SZ * H_SZ) return;
  float s0 = 0.f, s1 = 0.f, s2 = 0.f, s3 = 0.f;
  for (int sp = 0; sp < SPLITS; sp += 4) {       // SPLITS % 4 == 0
    s0 += part[(size_t)(sp + 0) * (B_SZ * H_SZ) + e];
    s1 += part[(size_t)(sp + 1) * (B_SZ * H_SZ) + e];
    s2 += part[(size_t)(sp + 2) * (B_SZ * H_SZ) + e];
    s3 += part[(size_t)(sp + 3) * (B_SZ * H_SZ) + e];
  }
  float s = b1[e % H_SZ] + ((s0 + s1) + (s2 + s3));
  y1[e] = fmaxf(s, 0.0f);
}

// ------------------------- K5: W2 + output heads ---------------------------
__global__ __launch_bounds__(256) void tfn_head(
    const float* __restrict__ y1, const float* __restrict__ W2,
    const float* __restrict__ b2,
    const float* __restrict__ Wo1, const float* __restrict__ bo1,
    const float* __restrict__ Wo2, const float* __restrict__ bo2,
    const float* __restrict__ Wo3, const float* __restrict__ bo3,
    float* __restrict__ out) {
  __shared__ float ys[B_SZ * H_SZ];
  __shared__ float fs[B_SZ * H_SZ];
  const int tid = threadIdx.x;

  for (int i = tid; i < B_SZ * H_SZ; i += 256) ys[i] = y1[i];
  __syncthreads();

  for (int i = tid; i < B_SZ * H_SZ; i += 256) {
    int b = i / H_SZ, h = i - b * H_SZ;
    float s0 = 0.f, s1 = 0.f, s2 = 0.f, s3 = 0.f;
    const float* yr = &ys[b * H_SZ];
    for (int k = 0; k < H_SZ; k += 4) {
      s0 = fmaf(yr[k + 0], W2[(k + 0) * H_SZ + h], s0);
      s1 = fmaf(yr[k + 1], W2[(k + 1) * H_SZ + h], s1);
      s2 = fmaf(yr[k + 2], W2[(k + 2) * H_SZ + h], s2);
      s3 = fmaf(yr[k + 3], W2[(k + 3) * H_SZ + h], s3);
    }
    float s = fmaxf(b2[h] + ((s0 + s1) + (s2 + s3)), 0.0f);
    fs[i] = s;
    out[i] = s;                                   // features -> d_out[0:6144)
  }
  __syncthreads();

  for (int i = tid; i < B_SZ * 6; i += 256) {     // emos
    int b = i / 6, o = i - b * 6;
    float s0 = 0.f, s1 = 0.f;
    const float* fr = &fs[b * H_SZ];
    for (int k = 0; k < H_SZ; k += 2) {
      s0 = fmaf(fr[k + 0], Wo1[(k + 0) * 6 + o], s0);
      s1 = fmaf(fr[k + 1], Wo1[(k + 1) * 6 + o], s1);
    }
    out[6144 + i] = bo1[o] + s0 + s1;
  }
  for (int i = tid; i < B_SZ; i += 256) {         // vals
    float s0 = 0.f, s1 = 0.f;
    const float* fr = &fs[i * H_SZ];
    for (int k = 0; k < H_SZ; k += 2) {
      s0 = fmaf(fr[k + 0], Wo2[k + 0], s0);
      s1 = fmaf(fr[k + 1], Wo2[k + 1], s1);
    }
    out[6528 + i] = bo2[0] + s0 + s1;
  }
  for (int i = tid; i < B_SZ * 3; i += 256) {     // sents
    int b = i / 3, o = i - b * 3;
    float s0 = 0.f, s1 = 0.f;
    const float* fr = &fs[b * H_SZ];
    for (int k = 0; k < H_SZ; k += 2) {
      s0 = fmaf(fr[k + 0], Wo3[(k + 0) * 3 + o], s0);
      s1 = fmaf(fr[k + 1], Wo3[(k + 1) * 3 + o], s1);
    }
    out[6592 + i] = bo3[o] + s0 + s1;
  }
  if (tid == 0) out[6784] = 0.0f;                 // interloss
}

// ------------------------------- launcher ----------------------------------
extern "C" void kernel_launch(void* const* d_in, const int* in_sizes, int n_in,
                              void* d_out, int out_size, void* d_ws, size_t ws_size,
                              hipStream_t stream) {
  const float* audios = (const float*)d_in[0];
  const float* texts  = (const float*)d_in[1];
  const float* videos = (const float*)d_in[2];
  const float* Wa  = (const float*)d_in[3];
  const float* ba  = (const float*)d_in[4];
  const float* Wt  = (const float*)d_in[5];
  const float* bt  = (const float*)d_in[6];
  const float* Wv  = (const float*)d_in[7];
  const float* bv  = (const float*)d_in[8];
  const float* W1  = (const float*)d_in[9];
  const float* b1  = (const float*)d_in[10];
  const float* W2  = (const float*)d_in[11];
  const float* b2  = (const float*)d_in[12];
  const float* Wo1 = (const float*)d_in[13];
  const float* bo1 = (const float*)d_in[14];
  const float* Wo2 = (const float*)d_in[15];
  const float* bo2 = (const float*)d_in[16];
  const float* Wo3 = (const float*)d_in[17];
  const float* bo3 = (const float*)d_in[18];
  float* out = (float*)d_out;

  // Workspace layout (floats), total ~15.1 MB.
  float* ws   = (float*)d_ws;
  float* a1   = ws;                                  // 64*97
  float* v1   = a1 + B_SZ * HP1;                     // 64*97
  float* t1   = v1 + B_SZ * HP1;                     // 64*97
  float* av   = t1 + B_SZ * HP1;                     // 64*97*97
  float* part = av + B_SZ * HP1 * HP1;               // SPLITS*64*96
  float* y1   = part + (size_t)SPLITS * B_SZ * H_SZ; // 64*96

  tfn_encode<<<(3 * B_SZ * HP1 + 255) / 256, 256, 0, stream>>>(
      audios, texts, videos, Wa, ba, Wt, bt, Wv, bv, a1, t1, v1);
  tfn_av<<<(B_SZ * HP1 * HP1 + 255) / 256, 256, 0, stream>>>(a1, v1, av);
  tfn_gemm<<<SPLITS, 256, 0, stream>>>(W1, av, t1, part);
  tfn_reduce<<<(B_SZ * H_SZ + 255) / 256, 256, 0, stream>>>(part, b1, y1);
  tfn_head<<<1, 256, 0, stream>>>(y1, W2, b2, Wo1, bo1, Wo2, bo2, Wo3, bo3, out);
}